// DMoN_89077621719556
// MI455X (gfx1250) — compile-verified
//
#include <hip/hip_runtime.h>
#include <math.h>

#define N_NODES 100000
#define N_FEAT  256
#define N_EDGES 3200000
#define NK      16          // clusters
#define COLLAPSE_REG 0.1f

typedef float v2f __attribute__((ext_vector_type(2)));
typedef float v8f __attribute__((ext_vector_type(8)));

// ---- workspace layout (float offsets) ----
#define WS_CS     0                     // cluster_sizes [16]
#define WS_AD     16                    // A^T d         [16]
#define WS_SUMVAL 32                    // sum(edge_val) = 2*n_edges
#define WS_TRACE  33                    // trace(graph_pooled)
#define WS_M      64                    // A^T F stored [d][k] = d*16+k  (256*16)
#define WS_DEG    (WS_M + N_FEAT*NK)    // degrees [100000]
#define WS_TOTAL  (WS_DEG + N_NODES)

// ---------------- pass 0: zero workspace ----------------
__global__ void dmon_zero_ws(float* __restrict__ ws) {
  int i = blockIdx.x * blockDim.x + threadIdx.x;
  if (i < WS_TOTAL) ws[i] = 0.0f;
}

// ---------------- pass 1: degrees + sum(edge_val) ----------------
__global__ void dmon_degrees(const int* __restrict__ erow,
                             const float* __restrict__ ev,
                             float* __restrict__ ws) {
  __shared__ float red[8];
  int tid = blockIdx.x * blockDim.x + threadIdx.x;
  int stride = gridDim.x * blockDim.x;
  float local = 0.0f;
  for (int i = tid; i < N_EDGES; i += stride) {
    float v = ev[i];
    atomicAdd(&ws[WS_DEG + erow[i]], v);
    local += v;
  }
  for (int off = 16; off > 0; off >>= 1) local += __shfl_down(local, off, 32);
  int lane = threadIdx.x & 31, wid = threadIdx.x >> 5;
  if (lane == 0) red[wid] = local;
  __syncthreads();
  if (threadIdx.x == 0) {
    float s = 0.0f;
    for (int w = 0; w < 8; ++w) s += red[w];
    atomicAdd(&ws[WS_SUMVAL], s);
  }
}

// ---------------- pass 2: fused node kernel ----------------
// one wave = one 16-node tile; block = 8 waves = 128 nodes
__global__ __launch_bounds__(256) void dmon_nodes(
    const float* __restrict__ feat, const float* __restrict__ Wm,
    const float* __restrict__ bvec, const float* __restrict__ deg,
    float* __restrict__ ws, float* __restrict__ assign_out) {
  __shared__ float ldsA[8][16 * 17];      // per-wave assignment tile (padded)
  __shared__ float ldsM[N_FEAT * NK];     // block accumulator for A^T F, [d][k]
  __shared__ float lds_cs[NK];
  __shared__ float lds_ad[NK];

  int t = threadIdx.x;
  for (int i = t; i < N_FEAT * NK; i += 256) ldsM[i] = 0.0f;
  if (t < NK) { lds_cs[t] = 0.0f; lds_ad[t] = 0.0f; }
  __syncthreads();

  int wid = t >> 5, lane = t & 31;
  int n  = lane & 15;        // column within 16-group (cluster / d-lane)
  int hf = lane >> 4;        // lane half (0 or 1)
  int tile = blockIdx.x * 8 + wid;
  bool active = (tile * 16 < N_NODES);    // N_NODES % 16 == 0, tiles are full

  if (active) {
    int row0 = tile * 16;

    // ---- GEMM1: logits = F_tile[16x256] @ W[256x16]  (fp32 WMMA, K stepped by 4)
    v8f acc = {0.f,0.f,0.f,0.f,0.f,0.f,0.f,0.f};
    const float* frow = feat + (size_t)(row0 + n) * N_FEAT; // A-frag row = lane&15
    for (int k0 = 0; k0 < N_FEAT; k0 += 4) {
      int kk = k0 + 2 * hf;
      v2f a; a.x = frow[kk];               a.y = frow[kk + 1];
      v2f b; b.x = Wm[kk * NK + n];        b.y = Wm[(kk + 1) * NK + n];
      acc = __builtin_amdgcn_wmma_f32_16x16x4_f32(false, a, false, b,
                                                  (short)0, acc, false, false);
    }

    // ---- bias + row softmax (row M = r + 8*hf lives across one 16-lane group)
    float bias = bvec[n];
    float ar[8];
#pragma unroll
    for (int r = 0; r < 8; ++r) {
      float x = acc[r] + bias;
      float m = x;
      for (int off = 1; off < 16; off <<= 1) m = fmaxf(m, __shfl_xor(m, off, 16));
      float e = expf(x - m);
      float s = e;
      for (int off = 1; off < 16; off <<= 1) s += __shfl_xor(s, off, 16);
      ar[r] = e / s;
    }

    // ---- store assignments (global) + tile (LDS)
#pragma unroll
    for (int r = 0; r < 8; ++r) {
      int m = r + 8 * hf;
      assign_out[(size_t)(row0 + m) * NK + n] = ar[r];
      ldsA[wid][m * 17 + n] = ar[r];
    }

    // ---- cluster sizes
    float cs = 0.0f;
#pragma unroll
    for (int r = 0; r < 8; ++r) cs += ar[r];
    cs += __shfl_xor(cs, 16, 32);
    if (hf == 0) atomicAdd(&lds_cs[n], cs);

    // ---- A^T d  (degrees broadcast from lanes 0..15)
    float ddv = deg[row0 + n];
    float ad = 0.0f;
#pragma unroll
    for (int r = 0; r < 8; ++r) {
      float w = __shfl(ddv, r + 8 * hf, 32);
      ad += w * ar[r];
    }
    ad += __shfl_xor(ad, 16, 32);
    if (hf == 0) atomicAdd(&lds_ad[n], ad);

    // ---- GEMM2: Mt[d,k] += (F^T)[dx16nodes] @ A_tile[16x16]  (K = nodes, fp32 WMMA)
    for (int d0 = 0; d0 < N_FEAT; d0 += 16) {
      v8f c2 = {0.f,0.f,0.f,0.f,0.f,0.f,0.f,0.f};
#pragma unroll
      for (int m0 = 0; m0 < 16; m0 += 4) {
        int mm = m0 + 2 * hf;
        v2f a;                               // A-frag = F^T : rows are d, K is node
        a.x = feat[(size_t)(row0 + mm) * N_FEAT + d0 + n];
        a.y = feat[(size_t)(row0 + mm + 1) * N_FEAT + d0 + n];
        v2f b;                               // B-frag = assignment tile
        b.x = ldsA[wid][mm * 17 + n];
        b.y = ldsA[wid][(mm + 1) * 17 + n];
        c2 = __builtin_amdgcn_wmma_f32_16x16x4_f32(false, a, false, b,
                                                   (short)0, c2, false, false);
      }
#pragma unroll
      for (int r = 0; r < 8; ++r) {
        int d = d0 + r + 8 * hf;             // D-frag row = d, col = cluster n
        atomicAdd(&ldsM[d * NK + n], c2[r]);
      }
    }
  }
  __syncthreads();

  // ---- flush block accumulators to global workspace
  for (int i = t; i < N_FEAT * NK; i += 256) atomicAdd(&ws[WS_M + i], ldsM[i]);
  if (t < NK) {
    atomicAdd(&ws[WS_CS + t], lds_cs[t]);
    atomicAdd(&ws[WS_AD + t], lds_ad[t]);
  }
}

// ---------------- pass 3: trace(graph_pooled) over edges ----------------
__global__ void dmon_trace(const int* __restrict__ erow, const int* __restrict__ ecol,
                           const float* __restrict__ ev, const float* __restrict__ assign,
                           float* __restrict__ ws) {
  __shared__ float red[8];
  int tid = blockIdx.x * blockDim.x + threadIdx.x;
  int stride = gridDim.x * blockDim.x;
  float local = 0.0f;
  for (int i = tid; i < N_EDGES; i += stride) {
    const float4* ra = (const float4*)(assign + (size_t)erow[i] * NK);
    const float4* rb = (const float4*)(assign + (size_t)ecol[i] * NK);
    float d = 0.0f;
#pragma unroll
    for (int j = 0; j < 4; ++j) {
      float4 x = ra[j], y = rb[j];
      d += x.x * y.x + x.y * y.y + x.z * y.z + x.w * y.w;
    }
    local += ev[i] * d;
  }
  for (int off = 16; off > 0; off >>= 1) local += __shfl_down(local, off, 32);
  int lane = threadIdx.x & 31, wid = threadIdx.x >> 5;
  if (lane == 0) red[wid] = local;
  __syncthreads();
  if (threadIdx.x == 0) {
    float s = 0.0f;
    for (int w = 0; w < 8; ++w) s += red[w];
    atomicAdd(&ws[WS_TRACE], s);
  }
}

// ---------------- pass 4: finalize ----------------
__global__ void dmon_finalize(const float* __restrict__ ws, float* __restrict__ out) {
  int t = threadIdx.x;
  const float scale = 1.0507009873554805f, alpha = 1.6732632423543772f;
  for (int i = t; i < NK * N_FEAT; i += 256) {         // out[k*256+d]
    int k = i >> 8, d = i & 255;
    float x = ws[WS_M + d * NK + k] / ws[WS_CS + k];   // selu(A^T F / cs)
    out[i] = (x > 0.0f) ? scale * x : scale * alpha * (expf(x) - 1.0f);
  }
  if (t == 0) {
    float S  = ws[WS_SUMVAL];                          // = 2 * n_edges
    float tg = ws[WS_TRACE];
    float sad2 = 0.0f, scs2 = 0.0f;
    for (int k = 0; k < NK; ++k) {
      float a = ws[WS_AD + k]; sad2 += a * a;
      float c = ws[WS_CS + k]; scs2 += c * c;
    }
    float spectral = -(tg - sad2 / S) / S;
    float collapse = COLLAPSE_REG * (sqrtf(scs2) / (float)N_NODES * 4.0f - 1.0f);
    size_t base = (size_t)NK * N_FEAT + (size_t)N_NODES * NK;
    out[base]     = spectral;
    out[base + 1] = collapse;
  }
}

extern "C" void kernel_launch(void* const* d_in, const int* in_sizes, int n_in,
                              void* d_out, int out_size, void* d_ws, size_t ws_size,
                              hipStream_t stream) {
  const float* feat = (const float*)d_in[0];
  const float* Wm   = (const float*)d_in[1];
  const float* bvec = (const float*)d_in[2];
  const int*   erow = (const int*)d_in[3];
  const int*   ecol = (const int*)d_in[4];
  const float* ev   = (const float*)d_in[5];
  float* out = (float*)d_out;
  float* ws  = (float*)d_ws;
  float* assign = out + NK * N_FEAT;   // assignments live directly in d_out

  dmon_zero_ws<<<(WS_TOTAL + 255) / 256, 256, 0, stream>>>(ws);
  dmon_degrees<<<2048, 256, 0, stream>>>(erow, ev, ws);
  int ntiles = N_NODES / 16;           // 6250, exact
  dmon_nodes<<<(ntiles + 7) / 8, 256, 0, stream>>>(feat, Wm, bvec,
                                                   ws + WS_DEG, ws, assign);
  dmon_trace<<<4096, 256, 0, stream>>>(erow, ecol, ev, assign, ws);
  dmon_finalize<<<1, 256, 0, stream>>>(ws, out);
}